// NNPytorchAD_3659312136356
// MI455X (gfx1250) — compile-verified
//
#include <hip/hip_runtime.h>
#include <cmath>

#define NSAMP 2048
#define DIN   16
#define HID   256
#define MOUT  4
#define HP    260   // padded LDS row pitch in floats (260 % 64 == 4 -> conflict-free strided reads)

typedef float v2f __attribute__((ext_vector_type(2)));
typedef float v8f __attribute__((ext_vector_type(8)));

__device__ __forceinline__ v8f wmma_f32_k4(v2f a, v2f b, v8f c) {
  // D = A(16x4) * B(4x16) + C, fp32 matrix pipe
  return __builtin_amdgcn_wmma_f32_16x16x4_f32(false, a, false, b, (short)0, c, false, false);
}

__device__ __forceinline__ v2f lds_b64(const float* p) {
  // 8-byte-aligned LDS load of two consecutive floats
  return *(const v2f*)p;
}

__launch_bounds__(512, 1)
__global__ void mlp_fwd_jac_hess(const float* __restrict__ x,
                                 const float* __restrict__ W0, const float* __restrict__ b0,
                                 const float* __restrict__ W1, const float* __restrict__ b1,
                                 const float* __restrict__ W2, const float* __restrict__ b2,
                                 const float* __restrict__ W3, const float* __restrict__ b3,
                                 float* __restrict__ f_out,
                                 float* __restrict__ df_out,
                                 float* __restrict__ d2f_out)
{
  extern __shared__ float sm[];
  float* Hs   = sm;                 // 256 x HP   Hessian state H[r=(d1*16+d2)][h]
  float* Js   = Hs + 256 * HP;      // 16  x HP   Jacobian state (holds raw Jz transiently)
  float* aA   = Js + 16 * HP;       // 256        activation
  float* sp   = aA + HID;           // 256        sigma'
  float* spp  = sp + HID;           // 256        sigma''
  float* zbuf = spp + HID;          // 2 x 256    z partial sums (lane-half reduction)

  const int tid  = threadIdx.x;
  const int lane = tid & 31;
  const int wave = tid >> 5;        // 0..15
  const int lm   = lane & 15;       // M / N index within tile
  const int hi   = lane >> 4;       // lane half
  const int kh   = hi << 1;         // K sub-offset {0,2}
  const int c0   = wave << 4;       // this wave's output-column tile
  const int s    = blockIdx.x;
  const float* xs = x + s * DIN;

  const v8f vzero = {};

  // ---------------- layer 0: z0 = x W0 + b0, seed J and H ----------------
  if (tid < HID) {
    float z = b0[tid];
    #pragma unroll
    for (int d = 0; d < DIN; ++d) z += xs[d] * W0[d * HID + tid];
    float t  = tanhf(z);
    float s1 = 1.0f - t * t;
    aA[tid]  = t;
    sp[tid]  = s1;
    spp[tid] = -2.0f * t * s1;
  }
  __syncthreads();
  for (int i = tid; i < DIN * HID; i += blockDim.x) {
    int d = i >> 8, c = i & 255;
    Js[d * HP + c] = sp[c] * W0[d * HID + c];
  }
  for (int i = tid; i < 256 * HID; i += blockDim.x) {
    int r = i >> 8, c = i & 255;
    Hs[r * HP + c] = spp[c] * W0[(r >> 4) * HID + c] * W0[(r & 15) * HID + c];
  }
  __syncthreads();

  // ---------------- hidden layers 1 and 2 ----------------
  for (int l = 0; l < 2; ++l) {
    const float* W  = (l == 0) ? W1 : W2;
    const float* bb = (l == 0) ? b1 : b2;

    // ===== half-pass 1: Hz panels 0..7 + Jz + z-partials, one shared B frag per k-step
    v8f hacc[8] = {};
    v8f jacc = vzero;
    float zp = 0.0f;
    #pragma unroll 2
    for (int k0 = 0; k0 < HID; k0 += 4) {
      v2f b;
      b[0] = W[(k0 + kh)     * HID + c0 + lm];
      b[1] = W[(k0 + kh + 1) * HID + c0 + lm];
      // batch all A fragments first so the ds loads issue back-to-back
      v2f ah[8];
      #pragma unroll
      for (int p = 0; p < 8; ++p)
        ah[p] = lds_b64(&Hs[(p * 16 + lm) * HP + k0 + kh]);
      v2f aj = lds_b64(&Js[lm * HP + k0 + kh]);
      v2f av = lds_b64(&aA[k0 + kh]);
      zp += av[0] * b[0] + av[1] * b[1];                    // fused z = a@W partials
      jacc = wmma_f32_k4(aj, b, jacc);                      // fused Jz = J@W
      #pragma unroll
      for (int p = 0; p < 8; ++p)
        hacc[p] = wmma_f32_k4(ah[p], b, hacc[p]);           // Hz = H@W, 8-way B reuse
    }
    zbuf[hi * HID + c0 + lm] = zp;
    __syncthreads();   // (1) old Js/aA and Hs rows 0..127 fully read; zbuf complete

    // stash raw Jz into Js; refresh activations
    #pragma unroll
    for (int i = 0; i < 8; ++i)
      Js[(hi * 8 + i) * HP + c0 + lm] = jacc[i];
    if (tid < HID) {
      float z  = zbuf[tid] + zbuf[HID + tid] + bb[tid];
      float t  = tanhf(z);
      float s1 = 1.0f - t * t;
      aA[tid]  = t;
      sp[tid]  = s1;
      spp[tid] = -2.0f * t * s1;
    }
    __syncthreads();   // (2) Jz + activations visible

    // in-place update of rows 0..127 (pass-2 reads rows 128..255: disjoint, no barrier)
    {
      const float spc  = sp[c0 + lm];
      const float sppc = spp[c0 + lm];
      #pragma unroll
      for (int p = 0; p < 8; ++p) {
        const float j1 = Js[p * HP + c0 + lm];              // Jz[d1=p][c]
        #pragma unroll
        for (int i = 0; i < 8; ++i) {
          int r = p * 16 + hi * 8 + i;
          Hs[r * HP + c0 + lm] = spc * hacc[p][i] + sppc * j1 * Js[(r & 15) * HP + c0 + lm];
        }
      }
    }

    // ===== half-pass 2: Hz panels 8..15
    #pragma unroll
    for (int p = 0; p < 8; ++p) hacc[p] = vzero;
    #pragma unroll 2
    for (int k0 = 0; k0 < HID; k0 += 4) {
      v2f b;
      b[0] = W[(k0 + kh)     * HID + c0 + lm];
      b[1] = W[(k0 + kh + 1) * HID + c0 + lm];
      v2f ah[8];
      #pragma unroll
      for (int p = 0; p < 8; ++p)
        ah[p] = lds_b64(&Hs[((p + 8) * 16 + lm) * HP + k0 + kh]);
      #pragma unroll
      for (int p = 0; p < 8; ++p)
        hacc[p] = wmma_f32_k4(ah[p], b, hacc[p]);
    }
    __syncthreads();   // (3) rows 128..255 fully read by every wave
    {
      const float spc  = sp[c0 + lm];
      const float sppc = spp[c0 + lm];
      #pragma unroll
      for (int p = 0; p < 8; ++p) {
        const float j1 = Js[(p + 8) * HP + c0 + lm];        // Jz[d1=p+8][c]
        #pragma unroll
        for (int i = 0; i < 8; ++i) {
          int r = (p + 8) * 16 + hi * 8 + i;
          Hs[r * HP + c0 + lm] = spc * hacc[p][i] + sppc * j1 * Js[(r & 15) * HP + c0 + lm];
        }
      }
    }
    __syncthreads();   // (4) pass-2 Jz reads complete

    // J' = sigma' ⊙ Jz, in place
    for (int i = tid; i < DIN * HID; i += blockDim.x) {
      int d = i >> 8, c = i & 255;
      Js[d * HP + c] *= sp[c];
    }
    __syncthreads();   // (5)
  }

  // ---------------- output layer (linear, W3: 256x4) ----------------
  // d2f = H @ W3 via WMMA: wave p owns 16-row panel p; N-tile cols 0..3 live, rest zero
  {
    v8f oacc = vzero;
    #pragma unroll 2
    for (int k0 = 0; k0 < HID; k0 += 4) {
      v2f b;
      b[0] = (lm < MOUT) ? W3[(k0 + kh)     * MOUT + lm] : 0.0f;
      b[1] = (lm < MOUT) ? W3[(k0 + kh + 1) * MOUT + lm] : 0.0f;
      v2f ah = lds_b64(&Hs[(wave * 16 + lm) * HP + k0 + kh]);
      oacc = wmma_f32_k4(ah, b, oacc);
    }
    if (lm < MOUT) {
      #pragma unroll
      for (int i = 0; i < 8; ++i) {
        int r = wave * 16 + hi * 8 + i;                     // r = d1*16 + d2
        d2f_out[s * 256 * MOUT + r * MOUT + lm] = oacc[i];  // (ns, d1, d2, m)
      }
    }
  }
  // df = J @ W3 via WMMA on wave 0
  if (wave == 0) {
    v8f dacc = vzero;
    #pragma unroll 2
    for (int k0 = 0; k0 < HID; k0 += 4) {
      v2f b;
      b[0] = (lm < MOUT) ? W3[(k0 + kh)     * MOUT + lm] : 0.0f;
      b[1] = (lm < MOUT) ? W3[(k0 + kh + 1) * MOUT + lm] : 0.0f;
      v2f aj = lds_b64(&Js[lm * HP + k0 + kh]);
      dacc = wmma_f32_k4(aj, b, dacc);
    }
    if (lm < MOUT) {
      #pragma unroll
      for (int i = 0; i < 8; ++i) {
        int d = hi * 8 + i;
        df_out[s * DIN * MOUT + d * MOUT + lm] = dacc[i];   // (ns, d, m)
      }
    }
  }
  // f = a @ W3 + b3 (tiny: 4 dots)
  if (wave == 1 && lane < MOUT) {
    float acc = b3[lane];
    for (int k = 0; k < HID; ++k) acc += aA[k] * W3[k * MOUT + lane];
    f_out[s * MOUT + lane] = acc;
  }
}

extern "C" void kernel_launch(void* const* d_in, const int* in_sizes, int n_in,
                              void* d_out, int out_size, void* d_ws, size_t ws_size,
                              hipStream_t stream) {
  (void)in_sizes; (void)n_in; (void)d_ws; (void)ws_size; (void)out_size;
  const float* x  = (const float*)d_in[0];
  const float* W0 = (const float*)d_in[1];
  const float* b0 = (const float*)d_in[2];
  const float* W1 = (const float*)d_in[3];
  const float* b1 = (const float*)d_in[4];
  const float* W2 = (const float*)d_in[5];
  const float* b2 = (const float*)d_in[6];
  const float* W3 = (const float*)d_in[7];
  const float* b3 = (const float*)d_in[8];

  float* out     = (float*)d_out;
  float* f_out   = out;                                        // 2048*4
  float* df_out  = out + NSAMP * MOUT;                         // 2048*16*4
  float* d2f_out = out + NSAMP * MOUT + NSAMP * DIN * MOUT;    // 2048*16*16*4

  const size_t shbytes = (size_t)(256 * HP + 16 * HP + 3 * HID + 2 * HID) * sizeof(float); // 288 KB
  hipFuncSetAttribute(reinterpret_cast<const void*>(mlp_fwd_jac_hess),
                      hipFuncAttributeMaxDynamicSharedMemorySize, (int)shbytes);

  mlp_fwd_jac_hess<<<NSAMP, 512, shbytes, stream>>>(x, W0, b0, W1, b1, W2, b2, W3, b3,
                                                    f_out, df_out, d2f_out);
}